// CausalSelfAttention_3384434229598
// MI455X (gfx1250) — compile-verified
//
#include <hip/hip_runtime.h>
#include <hip/hip_bf16.h>

typedef __attribute__((ext_vector_type(2))) float v2f;
typedef __attribute__((ext_vector_type(8))) float v8f;
typedef unsigned int u32x4 __attribute__((ext_vector_type(4)));
typedef int          i32x4 __attribute__((ext_vector_type(4)));
typedef int          i32x8 __attribute__((ext_vector_type(8)));

#define N_EMBD   1024
#define N_HEAD   16
#define HEAD_DIM 64
#define BATCH    4
#define SEQ      2048

// TDM-staged tile geometry for the GEMMs
#define KC        64                  // K chunk
#define ROWSTRIDE 68                  // 64 + 4 DWORD pad per row (bank-conflict-free)
#define A_ROWS    128
#define B_ROWS    256
#define A_ELT     (A_ROWS * ROWSTRIDE)   // floats per A buffer
#define B_ELT     (B_ROWS * ROWSTRIDE)   // floats per B buffer
#define LDS_FLOATS (2 * (A_ELT + B_ELT)) // double-buffered
#define LDS_BYTES  (LDS_FLOATS * 4)      // 208896 B <= 320 KB

__device__ __forceinline__ v8f wmma_f32(v2f a, v2f b, v8f c) {
  // D(16x16,f32) = A(16x4,f32) x B(4x16,f32) + C
  return __builtin_amdgcn_wmma_f32_16x16x4_f32(
      false, a, false, b, (short)0, c, false, false);
}

// -----------------------------------------------------------------------------
// TDM: async-load a 2D tile (tile_d0 x tile_d1, 4-byte elems) from a row-major
// tensor (row length tensor_d0, stride_d0, tensor_d1 rows) into LDS at lds_off,
// inserting 4 DWORDs of padding after every 64 DWORDs (row stride 68 floats).
// D# packing per CDNA5 ISA 8.3/8.4.  (6-arg toolchain builtin.)
// -----------------------------------------------------------------------------
__device__ __forceinline__ void tdm_load_2d(const float* gptr, unsigned lds_off,
                                            int tile_d0, int tile_d1,
                                            int tensor_d0, int tensor_d1,
                                            int stride_d0) {
  unsigned long long ga = (unsigned long long)(const void*)gptr;
  u32x4 g0;
  g0.x = 1u;                                           // count=1, user mode
  g0.y = lds_off;                                      // LDS byte address
  g0.z = (unsigned)(ga & 0xFFFFFFFFu);                 // global_addr[31:0]
  g0.w = (unsigned)((ga >> 32) & 0x1FFFFFFu)           // global_addr[56:32]
         | (2u << 30);                                 // type=2 ("image")
  i32x8 g1;
  g1[0] = (int)((2u << 16)                             // data_size = 4B
                | (1u << 20)                           // pad_enable
                | (5u << 22)                           // pad_interval: 64 DW
                | (3u << 25));                         // pad_amount: 4 DW
  g1[1] = (int)(((unsigned)tensor_d0 & 0xFFFFu) << 16);          // dim0[15:0]
  g1[2] = (int)((((unsigned)tensor_d0 >> 16) & 0xFFFFu)          // dim0[31:16]
                | (((unsigned)tensor_d1 & 0xFFFFu) << 16));      // dim1[15:0]
  g1[3] = (int)((((unsigned)tensor_d1 >> 16) & 0xFFFFu)          // dim1[31:16]
                | ((unsigned)tile_d0 << 16));                    // tile_dim0
  g1[4] = (int)((unsigned)tile_d1);                    // tile_dim1 (tile_dim2=0)
  g1[5] = stride_d0;                                   // dim0_stride[31:0]
  g1[6] = 0;                                           // stride hi / dim1_stride
  g1[7] = 0;
  i32x4 z4 = {0, 0, 0, 0};                             // groups 2/3: 2-D tile
  i32x8 z8 = {0, 0, 0, 0, 0, 0, 0, 0};                 // extra group (unused)
  __builtin_amdgcn_tensor_load_to_lds(g0, g1, z4, z4, z8, 0);
}

// -----------------------------------------------------------------------------
// Y[M,N] = X[M,K] @ W[N,K]^T + bias[N]
// block = 256 threads (8 waves); wave tile 64x64; block tile 128x256.
// X/W tiles streamed global->LDS by the Tensor Data Mover, double-buffered;
// fragments read from LDS with ds_load_b64 (conflict-free via row pad).
// Requires M % 128 == 0, N % 256 == 0, K % 64 == 0.
// -----------------------------------------------------------------------------
__global__ __launch_bounds__(256) void gemm_xwt(
    const float* __restrict__ X, const float* __restrict__ W,
    const float* __restrict__ bias, float* __restrict__ Y,
    int M, int N, int K) {
  extern __shared__ __align__(16) float smem[];

  const int lane = threadIdx.x & 31;
  const int wave = threadIdx.x >> 5;
  const int lm = lane & 15;          // row/col within 16
  const int kh = (lane >> 4) & 1;    // K half select (K: 0,1 vs 2,3)
  const int wm = wave >> 2;          // 0..1
  const int wn = wave & 3;           // 0..3

  const int bm0 = blockIdx.x * 128;
  const int bn0 = blockIdx.y * 256;
  const int m0 = bm0 + wm * 64;
  const int n0 = bn0 + wn * 64;

  v8f acc[4][4];
#pragma unroll
  for (int mt = 0; mt < 4; ++mt)
#pragma unroll
    for (int nt = 0; nt < 4; ++nt)
#pragma unroll
      for (int r = 0; r < 8; ++r) acc[mt][nt][r] = 0.0f;

  const unsigned ldsBase = (unsigned)(unsigned long long)(void*)smem;
  const unsigned ldsA[2] = {ldsBase, ldsBase + A_ELT * 4u};
  const unsigned ldsB[2] = {ldsBase + 2u * A_ELT * 4u,
                            ldsBase + 2u * A_ELT * 4u + B_ELT * 4u};

  const int chunks = K / KC;

  // prologue: stream chunk 0
  if (wave == 0)
    tdm_load_2d(X + (size_t)bm0 * K, ldsA[0], KC, A_ROWS, K, M, K);
  if (wave == 1)
    tdm_load_2d(W + (size_t)bn0 * K, ldsB[0], KC, B_ROWS, K, N, K);
  if (wave < 2) __builtin_amdgcn_s_wait_tensorcnt(0);
  __syncthreads();

  for (int c = 0; c < chunks; ++c) {
    // kick off next chunk's DMA while computing on the current one
    if (c + 1 < chunks) {
      int nb = (c + 1) & 1;
      if (wave == 0)
        tdm_load_2d(X + (size_t)bm0 * K + (c + 1) * KC, ldsA[nb], KC, A_ROWS,
                    K, M, K);
      if (wave == 1)
        tdm_load_2d(W + (size_t)bn0 * K + (c + 1) * KC, ldsB[nb], KC, B_ROWS,
                    K, N, K);
    }

    const float* Ab = smem + (c & 1) * A_ELT;
    const float* Bb = smem + 2 * A_ELT + (c & 1) * B_ELT;

    const float* ap0 = Ab + (wm * 64 + 0  + lm) * ROWSTRIDE + 2 * kh;
    const float* ap1 = Ab + (wm * 64 + 16 + lm) * ROWSTRIDE + 2 * kh;
    const float* ap2 = Ab + (wm * 64 + 32 + lm) * ROWSTRIDE + 2 * kh;
    const float* ap3 = Ab + (wm * 64 + 48 + lm) * ROWSTRIDE + 2 * kh;
    const float* bp0 = Bb + (wn * 64 + 0  + lm) * ROWSTRIDE + 2 * kh;
    const float* bp1 = Bb + (wn * 64 + 16 + lm) * ROWSTRIDE + 2 * kh;
    const float* bp2 = Bb + (wn * 64 + 32 + lm) * ROWSTRIDE + 2 * kh;
    const float* bp3 = Bb + (wn * 64 + 48 + lm) * ROWSTRIDE + 2 * kh;

#pragma unroll
    for (int kk = 0; kk < KC; kk += 4) {
      v2f a0 = *(const v2f*)(ap0 + kk);
      v2f a1 = *(const v2f*)(ap1 + kk);
      v2f a2 = *(const v2f*)(ap2 + kk);
      v2f a3 = *(const v2f*)(ap3 + kk);
      v2f b0 = *(const v2f*)(bp0 + kk);
      v2f b1 = *(const v2f*)(bp1 + kk);
      v2f b2 = *(const v2f*)(bp2 + kk);
      v2f b3 = *(const v2f*)(bp3 + kk);

      acc[0][0] = wmma_f32(a0, b0, acc[0][0]);
      acc[0][1] = wmma_f32(a0, b1, acc[0][1]);
      acc[0][2] = wmma_f32(a0, b2, acc[0][2]);
      acc[0][3] = wmma_f32(a0, b3, acc[0][3]);
      acc[1][0] = wmma_f32(a1, b0, acc[1][0]);
      acc[1][1] = wmma_f32(a1, b1, acc[1][1]);
      acc[1][2] = wmma_f32(a1, b2, acc[1][2]);
      acc[1][3] = wmma_f32(a1, b3, acc[1][3]);
      acc[2][0] = wmma_f32(a2, b0, acc[2][0]);
      acc[2][1] = wmma_f32(a2, b1, acc[2][1]);
      acc[2][2] = wmma_f32(a2, b2, acc[2][2]);
      acc[2][3] = wmma_f32(a2, b3, acc[2][3]);
      acc[3][0] = wmma_f32(a3, b0, acc[3][0]);
      acc[3][1] = wmma_f32(a3, b1, acc[3][1]);
      acc[3][2] = wmma_f32(a3, b2, acc[3][2]);
      acc[3][3] = wmma_f32(a3, b3, acc[3][3]);
    }

    if (c + 1 < chunks && wave < 2) __builtin_amdgcn_s_wait_tensorcnt(0);
    __syncthreads();
  }

  float bv[4];
#pragma unroll
  for (int nt = 0; nt < 4; ++nt) bv[nt] = bias[n0 + nt * 16 + lm];

#pragma unroll
  for (int mt = 0; mt < 4; ++mt) {
#pragma unroll
    for (int nt = 0; nt < 4; ++nt) {
#pragma unroll
      for (int r = 0; r < 8; ++r) {
        // C/D layout: vgpr r holds row (r + 8*kh), col = lm
        int row = m0 + mt * 16 + r + 8 * kh;
        int col = n0 + nt * 16 + lm;
        Y[(size_t)row * N + col] = acc[mt][nt][r] + bv[nt];
      }
    }
  }
}

// -----------------------------------------------------------------------------
// Flash attention, fp32 WMMA. One wave handles TWO adjacent 16-query tiles of
// one (b,h): K/V fragments loaded once, reused for both tiles.
// qkv layout: [(b*T + t) * 3C + n], q at n, k at C+n, v at 2C+n (n = h*D + d).
// -----------------------------------------------------------------------------
__global__ __launch_bounds__(256) void attn_fwd(
    const float* __restrict__ qkv, float* __restrict__ out) {
  const int C  = N_EMBD;
  const int C3 = 3 * N_EMBD;
  const int T  = SEQ;
  const int H  = N_HEAD;
  const int D  = HEAD_DIM;
  const int QT = T / 16;
  const int QP = QT / 2;

  const int lane = threadIdx.x & 31;
  const int wave = threadIdx.x >> 5;
  const int lm = lane & 15;
  const int kh = (lane >> 4) & 1;

  int idx = blockIdx.x * 8 + wave;
  const int p  = idx % QP;
  const int h  = (idx / QP) % H;
  const int b  = idx / (QP * H);
  const int qt0 = 2 * p;
  const int qt1 = 2 * p + 1;

  const float scale = rsqrtf((float)D);

  v2f qa[2][16];
#pragma unroll
  for (int qi = 0; qi < 2; ++qi) {
    const float* qbase =
        qkv + (size_t)(b * T + (qt0 + qi) * 16 + lm) * C3 + h * D + 2 * kh;
#pragma unroll
    for (int ks = 0; ks < 16; ++ks) {
      v2f t = *(const v2f*)(qbase + 4 * ks);
      qa[qi][ks] = t * scale;
    }
  }

  v8f o[2][4];
  float mrow[2][8], lrow[2][8];
#pragma unroll
  for (int qi = 0; qi < 2; ++qi) {
#pragma unroll
    for (int nt = 0; nt < 4; ++nt)
#pragma unroll
      for (int r = 0; r < 8; ++r) o[qi][nt][r] = 0.0f;
#pragma unroll
    for (int r = 0; r < 8; ++r) {
      mrow[qi][r] = -__builtin_inff();
      lrow[qi][r] = 0.0f;
    }
  }

  __shared__ float pbuf[8][2][256];
  float* myp = &pbuf[wave][0][0];

  for (int kt = 0; kt <= qt1; ++kt) {
    v2f kb[16];
    {
      const float* kbase =
          qkv + (size_t)(b * T + kt * 16 + lm) * C3 + C + h * D + 2 * kh;
#pragma unroll
      for (int ks = 0; ks < 16; ++ks)
        kb[ks] = *(const v2f*)(kbase + 4 * ks);
    }

    const int qlo = (kt <= qt0) ? 0 : 1;   // wave-uniform

    for (int qi = qlo; qi < 2; ++qi) {
      v8f s;
#pragma unroll
      for (int r = 0; r < 8; ++r) s[r] = 0.0f;
#pragma unroll
      for (int ks = 0; ks < 16; ++ks) s = wmma_f32(qa[qi][ks], kb[ks], s);

      if (kt == qt0 + qi) {
#pragma unroll
        for (int r = 0; r < 8; ++r)
          if (lm > r + 8 * kh) s[r] = -__builtin_inff();
      }

#pragma unroll
      for (int r = 0; r < 8; ++r) {
        float v = s[r];
        v = fmaxf(v, __shfl_xor(v, 1, 32));
        v = fmaxf(v, __shfl_xor(v, 2, 32));
        v = fmaxf(v, __shfl_xor(v, 4, 32));
        v = fmaxf(v, __shfl_xor(v, 8, 32));
        float mnew  = fmaxf(mrow[qi][r], v);
        float alpha = __expf(mrow[qi][r] - mnew);
        mrow[qi][r] = mnew;
        float pv = __expf(s[r] - mnew);
        s[r] = pv;
        float rs = pv;
        rs += __shfl_xor(rs, 1, 32);
        rs += __shfl_xor(rs, 2, 32);
        rs += __shfl_xor(rs, 4, 32);
        rs += __shfl_xor(rs, 8, 32);
        lrow[qi][r] = lrow[qi][r] * alpha + rs;
#pragma unroll
        for (int nt = 0; nt < 4; ++nt) o[qi][nt][r] *= alpha;
      }

#pragma unroll
      for (int r = 0; r < 8; ++r)
        myp[qi * 256 + (r + 8 * kh) * 16 + lm] = s[r];
    }
    asm volatile("s_wait_dscnt 0" ::: "memory");

    {
      const float* vbase = qkv + (size_t)(b * T + kt * 16) * C3 + 2 * C + h * D;
#pragma unroll
      for (int ks = 0; ks < 4; ++ks) {
        int krow = 4 * ks + 2 * kh;
        v2f pa1 = *(const v2f*)(myp + 256 + lm * 16 + 4 * ks + 2 * kh);
        v2f pa0 = *(const v2f*)(myp +       lm * 16 + 4 * ks + 2 * kh);
#pragma unroll
        for (int nt = 0; nt < 4; ++nt) {
          v2f vb;
          vb.x = vbase[(size_t)(krow)     * C3 + nt * 16 + lm];
          vb.y = vbase[(size_t)(krow + 1) * C3 + nt * 16 + lm];
          o[1][nt] = wmma_f32(pa1, vb, o[1][nt]);
          if (qlo == 0) o[0][nt] = wmma_f32(pa0, vb, o[0][nt]);
        }
      }
    }
    asm volatile("" ::: "memory");
  }

#pragma unroll
  for (int qi = 0; qi < 2; ++qi) {
    float inv[8];
#pragma unroll
    for (int r = 0; r < 8; ++r) inv[r] = 1.0f / lrow[qi][r];
#pragma unroll
    for (int nt = 0; nt < 4; ++nt) {
#pragma unroll
      for (int r = 0; r < 8; ++r) {
        int row = (qt0 + qi) * 16 + r + 8 * kh;
        out[(size_t)(b * T + row) * C + h * D + nt * 16 + lm] =
            o[qi][nt][r] * inv[r];
      }
    }
  }
}

// -----------------------------------------------------------------------------
extern "C" void kernel_launch(void* const* d_in, const int* in_sizes, int n_in,
                              void* d_out, int out_size, void* d_ws, size_t ws_size,
                              hipStream_t stream) {
  const float* x      = (const float*)d_in[0];   // (B,T,C)
  const float* w_attn = (const float*)d_in[1];   // (3C,C)
  const float* b_attn = (const float*)d_in[2];   // (3C,)
  const float* w_proj = (const float*)d_in[3];   // (C,C)
  const float* b_proj = (const float*)d_in[4];   // (C,)
  float* out = (float*)d_out;                    // (B,T,C)

  const int M  = BATCH * SEQ;      // 8192
  const int C  = N_EMBD;           // 1024
  const int C3 = 3 * N_EMBD;       // 3072

  float* qkv  = (float*)d_ws;                    // 96 MB
  float* atto = qkv + (size_t)M * C3;            // 32 MB

  dim3 blk(256);

  // 1) qkv = x @ w_attn^T + b_attn   (TDM-staged GEMM)
  gemm_xwt<<<dim3(M / 128, C3 / 256), blk, LDS_BYTES, stream>>>(
      x, w_attn, b_attn, qkv, M, C3, C);
  // 2) flash attention (2 query tiles per wave), merged-head output
  {
    int pairs = BATCH * N_HEAD * (SEQ / 32);     // 4096 waves
    attn_fwd<<<dim3(pairs / 8), blk, 0, stream>>>(qkv, atto);
  }
  // 3) out = atto @ w_proj^T + b_proj
  gemm_xwt<<<dim3(M / 128, C / 256), blk, LDS_BYTES, stream>>>(
      atto, w_proj, b_proj, out, M, C, C);
}